// HabanaAttentionImpl_89902255440424
// MI455X (gfx1250) — compile-verified
//
#include <hip/hip_runtime.h>
#include <math.h>

// ---- problem constants (match reference setup_inputs) ----
#define B_SZ   32
#define H_SZ   32
#define KVH    8
#define G      4          // H / KVH
#define D_SZ   128
#define BS     128        // tokens per cache block
#define MAXBLK 32
#define NWAVES 8
#define TILE   16         // tokens per wave-tile
#define KROW   132        // K tile row stride (floats): 528B, 16B multiple, conflict-free b64 reads
#define VROW   136        // V tile row stride (floats): 544B, 16B multiple, conflict-free 2addr reads
#define PROW   18         // P tile row stride (floats): 72B
#define SCALE  0.08838834764831845f

typedef __attribute__((ext_vector_type(2))) float f32x2;
typedef __attribute__((ext_vector_type(8))) float f32x8;

#define KTILE_F (16 * KROW)   // 2112 floats per K tile
#define VTILE_F (16 * VROW)   // 2176 floats per V tile

// dynamic LDS layout (floats):
//  qs  : 16*KROW                       = 2112   (rows 4..15 zero -> dead WMMA A rows)
//  ks  : NWAVES * 2 * KTILE_F          = 33792  (per-wave double-buffered K tiles)
//  vs  : NWAVES * 2 * VTILE_F          = 34816  (per-wave double-buffered V tiles)
//  ps  : NWAVES * 16*PROW              = 2304   (per-wave P tile, rows 4..15 zero)
//  red : NWAVES * G * D_SZ             = 4096
//  ml  : NWAVES * G * 2                = 64
#define LDS_FLOATS (16*KROW + NWAVES*2*KTILE_F + NWAVES*2*VTILE_F + NWAVES*16*PROW + NWAVES*G*D_SZ + NWAVES*G*2)

// Async-stage one 16-token tile of K and V into per-wave LDS buffers.
// One global_load_async_to_lds_b128 per row = 512B fully coalesced, 32 issues total,
// tracked on ASYNCcnt (no VGPR data, no loadcnt stalls).
__device__ __forceinline__ void stage_tile_async(const float* kbase, const float* vbase,
                                                 unsigned ksOff, unsigned vsOff, int lane)
{
    #pragma unroll
    for (int r = 0; r < 16; ++r) {
        unsigned            kd = ksOff + (unsigned)(r * KROW * 4 + lane * 16);
        unsigned long long  ka = (unsigned long long)(const void*)(kbase + (size_t)r * (KVH * D_SZ) + lane * 4);
        asm volatile("global_load_async_to_lds_b128 %0, %1, off" :: "v"(kd), "v"(ka) : "memory");
    }
    #pragma unroll
    for (int r = 0; r < 16; ++r) {
        unsigned            vd = vsOff + (unsigned)(r * VROW * 4 + lane * 16);
        unsigned long long  va = (unsigned long long)(const void*)(vbase + (size_t)r * (KVH * D_SZ) + lane * 4);
        asm volatile("global_load_async_to_lds_b128 %0, %1, off" :: "v"(vd), "v"(va) : "memory");
    }
}

__global__ void paged_attn_wmma_f32(const float* __restrict__ q,
                                    const float* __restrict__ kc,
                                    const float* __restrict__ vc,
                                    const int*   __restrict__ bt,
                                    const int*   __restrict__ cl,
                                    float*       __restrict__ out)
{
    extern __shared__ float smem[];
    float* qs    = smem;
    float* ksAll = qs    + 16 * KROW;
    float* vsAll = ksAll + NWAVES * 2 * KTILE_F;
    float* psAll = vsAll + NWAVES * 2 * VTILE_F;
    float* red   = psAll + NWAVES * 16 * PROW;
    float* ml    = red   + NWAVES * G * D_SZ;

    const int kvh  = blockIdx.x;
    const int b    = blockIdx.y;
    const int tid  = threadIdx.x;
    const int wave = tid >> 5;
    const int lane = tid & 31;
    const int row  = lane & 15;     // M or N index within the 16-wide stripe
    const int hi   = lane >> 4;     // selects K={2,3} half of the 16x4 / 4x16 operand

    float* ksW = ksAll + wave * 2 * KTILE_F;   // [2][KTILE_F]
    float* vsW = vsAll + wave * 2 * VTILE_F;   // [2][VTILE_F]
    float* ps  = psAll + wave * 16 * PROW;

    // Zero q rows (4..15 must stay zero) and the whole P region (rows 4..15 stay zero).
    for (int i = tid; i < 16 * KROW; i += 256)          qs[i] = 0.0f;
    for (int i = tid; i < NWAVES * 16 * PROW; i += 256) psAll[i] = 0.0f;
    __syncthreads();

    // Stage scaled q (G rows of D) into LDS.
    for (int i = tid; i < G * D_SZ; i += 256) {
        int g = i >> 7, d = i & (D_SZ - 1);
        qs[g * KROW + d] = q[((size_t)b * H_SZ + kvh * G + g) * D_SZ + d] * SCALE;
    }
    __syncthreads();

    const int ctx    = cl[b];
    const int ntiles = (ctx + TILE - 1) / TILE;

    float m[4] = {-3.0e38f, -3.0e38f, -3.0e38f, -3.0e38f};
    float l[4] = {0.f, 0.f, 0.f, 0.f};
    f32x8 acc[8] = {};   // PV accumulators: component g = query row, lane = dim 16t+lane

    int tile = wave;
    int buf  = 0;
    if (tile < ntiles) {
        const int    pb = bt[b * MAXBLK + (tile >> 3)];
        const size_t tb = ((size_t)(pb * BS + (tile & 7) * TILE) * KVH + kvh) * D_SZ;
        stage_tile_async(kc + tb, vc + tb,
                         (unsigned)(size_t)(void*)ksW, (unsigned)(size_t)(void*)vsW, lane);
    }

    for (; tile < ntiles; tile += NWAVES, buf ^= 1) {
        // Prefetch next tile into the other buffer, then retire current buffer's 32 loads.
        const int nxt = tile + NWAVES;
        if (nxt < ntiles) {
            const int    pb = bt[b * MAXBLK + (nxt >> 3)];
            const size_t tb = ((size_t)(pb * BS + (nxt & 7) * TILE) * KVH + kvh) * D_SZ;
            stage_tile_async(kc + tb, vc + tb,
                             (unsigned)(size_t)(void*)(ksW + (buf ^ 1) * KTILE_F),
                             (unsigned)(size_t)(void*)(vsW + (buf ^ 1) * VTILE_F), lane);
            asm volatile("s_wait_asynccnt 0x20" ::: "memory");  // current buffer done (in-order)
        } else {
            asm volatile("s_wait_asynccnt 0x0" ::: "memory");
        }
        const float* ks = ksW + buf * KTILE_F;
        const float* vs = vsW + buf * VTILE_F;

        // --- scores: C[q][tok] += sum_d q[q][d] * K[tok][d], 32 x WMMA f32 16x16x4 ---
        f32x8 c = {};
        #pragma unroll
        for (int k = 0; k < 32; ++k) {
            const int koff = 4 * k + 2 * hi;
            f32x2 a  = *(const f32x2*)(qs + row * KROW + koff);   // A[M=row][K pair]
            f32x2 kb = *(const f32x2*)(ks + row * KROW + koff);   // B[K pair][N=row(token)]
            c = __builtin_amdgcn_wmma_f32_16x16x4_f32(false, a, false, kb,
                                                      (short)0, c, false, false);
        }

        // --- online softmax over this tile (rows 0..3 valid, lanes 0..15 = tokens) ---
        const bool valid = (tile * TILE + row) < ctx;
        #pragma unroll
        for (int g = 0; g < 4; ++g) {
            float s = valid ? c[g] : -3.0e38f;
            float tmax = s;
            tmax = fmaxf(tmax, __shfl_xor(tmax, 1, 16));
            tmax = fmaxf(tmax, __shfl_xor(tmax, 2, 16));
            tmax = fmaxf(tmax, __shfl_xor(tmax, 4, 16));
            tmax = fmaxf(tmax, __shfl_xor(tmax, 8, 16));
            const float mnew = fmaxf(m[g], tmax);
            const float scl  = __expf(m[g] - mnew);
            const float pv   = __expf(s - mnew);
            float psum = pv;
            psum += __shfl_xor(psum, 1, 16);
            psum += __shfl_xor(psum, 2, 16);
            psum += __shfl_xor(psum, 4, 16);
            psum += __shfl_xor(psum, 8, 16);
            l[g] = l[g] * scl + psum;
            m[g] = mnew;
            #pragma unroll
            for (int t = 0; t < 8; ++t) acc[t][g] *= scl;   // rescale old accumulator
            if (hi == 0) ps[g * PROW + row] = pv;           // re-stripe P via LDS
        }

        // --- P·V: C[q][dim] += sum_tok P[q][tok] * V[tok][dim], 4x8 WMMA 16x16x4 ---
        #pragma unroll
        for (int kk = 0; kk < 4; ++kk) {
            const int tok = 4 * kk + 2 * hi;
            f32x2 a = *(const f32x2*)(ps + row * PROW + tok);     // A[M=row][K=tok pair]
            #pragma unroll
            for (int t = 0; t < 8; ++t) {
                f32x2 bv;                                          // B[K][N=dim 16t+row]
                bv.x = vs[tok * VROW + t * 16 + row];              // -> ds_load_2addr_b32
                bv.y = vs[(tok + 1) * VROW + t * 16 + row];
                acc[t] = __builtin_amdgcn_wmma_f32_16x16x4_f32(false, a, false, bv,
                                                               (short)0, acc[t], false, false);
            }
        }
    }

    // --- per-wave partials to LDS ---
    if (hi == 0) {
        #pragma unroll
        for (int t = 0; t < 8; ++t)
            #pragma unroll
            for (int g = 0; g < 4; ++g)
                red[(wave * G + g) * D_SZ + t * 16 + row] = acc[t][g];
        if (lane == 0) {
            #pragma unroll
            for (int g = 0; g < 4; ++g) {
                ml[(wave * G + g) * 2 + 0] = m[g];
                ml[(wave * G + g) * 2 + 1] = l[g];
            }
        }
    }
    __syncthreads();

    // --- cross-wave softmax merge + writeout ---
    for (int i = tid; i < G * D_SZ; i += 256) {
        const int g = i >> 7, d = i & (D_SZ - 1);
        float M = -3.0e38f;
        for (int w = 0; w < NWAVES; ++w) M = fmaxf(M, ml[(w * G + g) * 2]);
        float num = 0.f, den = 0.f;
        for (int w = 0; w < NWAVES; ++w) {
            const float e = __expf(ml[(w * G + g) * 2] - M);
            den += e * ml[(w * G + g) * 2 + 1];
            num += e * red[(w * G + g) * D_SZ + d];
        }
        out[((size_t)b * H_SZ + kvh * G + g) * D_SZ + d] = num / den;
    }
}

extern "C" void kernel_launch(void* const* d_in, const int* in_sizes, int n_in,
                              void* d_out, int out_size, void* d_ws, size_t ws_size,
                              hipStream_t stream) {
    const float* q  = (const float*)d_in[0];
    const float* kc = (const float*)d_in[1];
    const float* vc = (const float*)d_in[2];
    const int*   bt = (const int*)d_in[3];
    const int*   cl = (const int*)d_in[4];
    float* out = (float*)d_out;

    dim3 grid(KVH, B_SZ);
    dim3 block(256);
    size_t shmem = (size_t)LDS_FLOATS * sizeof(float);   // ~302 KB, under 320 KB/WGP
    paged_attn_wmma_f32<<<grid, block, shmem, stream>>>(q, kc, vc, bt, cl, out);
}